// LightGCN_89902255440890
// MI455X (gfx1250) — compile-verified
//
#include <hip/hip_runtime.h>
#include <hip/hip_bf16.h>
#include <stdint.h>

// ---------------------------------------------------------------------------
// LightGCN on MI455X (gfx1250): 3 x SpMM (COO, atomic scatter) + layer mean.
// SpMM core: wave-per-64-edges, lanes cover D=64 as float2, edge metadata
// double-buffered into LDS via async global->LDS copies (ASYNCcnt path).
// Final mean pass: x0.25 done with v_wmma_f32_16x16x4_f32 per 16x16 tile.
// ---------------------------------------------------------------------------

namespace {
constexpr int kUsers   = 100000;
constexpr int kItems   = 50000;
constexpr int kN       = kUsers + kItems;   // 150000
constexpr int kD       = 64;
constexpr int kNNZ     = 4800000;
constexpr int kLayers  = 3;
constexpr int kChunk   = 512;               // edges per staged chunk (NNZ % kChunk == 0)
constexpr int kNChunks = kNNZ / kChunk;     // 9375
constexpr int kTiles   = (kN / 16) * (kD / 16); // 9375 * 4 = 37500
constexpr int kElems   = kN * kD;           // 9,600,000 floats
}

// ---------------- CDNA5 feature probes ----------------
#if __has_builtin(__builtin_amdgcn_global_load_async_to_lds_b64)
#  define HAVE_ASYNC_LDS 1
#endif
#if __has_builtin(__builtin_amdgcn_s_wait_asynccnt)
#  define HAVE_WAIT_ASYNC 1
#endif
#if __has_builtin(__builtin_amdgcn_wmma_f32_16x16x4_f32)
#  define HAVE_WMMA_F32_4 1
#endif

typedef __attribute__((ext_vector_type(2))) int v2i;

__device__ __forceinline__ void async_cp_b64(void* lds_dst, const void* gsrc) {
#if defined(HAVE_ASYNC_LDS)
  __builtin_amdgcn_global_load_async_to_lds_b64(
      (__attribute__((address_space(1))) v2i*)(gsrc),
      (__attribute__((address_space(3))) v2i*)(lds_dst),
      0 /*imm offset*/, 0 /*cpol*/);
#else
  *(uint2*)lds_dst = *(const uint2*)gsrc;   // synchronous fallback
#endif
}

__device__ __forceinline__ void wait_async_all() {
#if defined(HAVE_WAIT_ASYNC)
  __builtin_amdgcn_s_wait_asynccnt(0);
#elif defined(HAVE_ASYNC_LDS)
  asm volatile("s_wait_asynccnt 0" ::: "memory");
#endif
}

// ---------------- elementwise kernels ----------------
__global__ void __launch_bounds__(256) init_kernel(
    const float4* __restrict__ ue, const float4* __restrict__ ie,
    float4* __restrict__ xcur, float4* __restrict__ acc) {
  const int n4 = kElems / 4;
  const int u4 = kUsers * kD / 4;
  for (int i = blockIdx.x * blockDim.x + threadIdx.x; i < n4;
       i += gridDim.x * blockDim.x) {
    float4 v = (i < u4) ? ue[i] : ie[i - u4];
    xcur[i] = v;
    acc[i]  = v;
  }
}

__global__ void __launch_bounds__(256) zero_kernel(float4* __restrict__ p) {
  const int n4 = kElems / 4;
  const float4 z = make_float4(0.f, 0.f, 0.f, 0.f);
  for (int i = blockIdx.x * blockDim.x + threadIdx.x; i < n4;
       i += gridDim.x * blockDim.x)
    p[i] = z;
}

__global__ void __launch_bounds__(256) acc_add_kernel(
    float4* __restrict__ acc, const float4* __restrict__ xn) {
  const int n4 = kElems / 4;
  for (int i = blockIdx.x * blockDim.x + threadIdx.x; i < n4;
       i += gridDim.x * blockDim.x) {
    float4 a = acc[i];
    const float4 b = xn[i];
    a.x += b.x; a.y += b.y; a.z += b.z; a.w += b.w;
    acc[i] = a;
  }
}

// ---------------- SpMM: y = A * x (COO, atomic scatter) ----------------
// Block = 256 threads = 8 waves. Each block streams chunks of 512 edges,
// metadata double-buffered into LDS with async copies. Each wave handles
// 64 edges/chunk; per edge the 32 lanes gather x[col][0:64) as float2 and
// atomically add val*x into y[row][0:64).
__global__ void __launch_bounds__(256) spmm_kernel(
    const int* __restrict__ rows, const int* __restrict__ cols,
    const float* __restrict__ vals, const float* __restrict__ x,
    float* __restrict__ y) {
  __shared__ __align__(16) int   s_r[2][kChunk];
  __shared__ __align__(16) int   s_c[2][kChunk];
  __shared__ __align__(16) float s_v[2][kChunk];

  const int t    = threadIdx.x;      // 0..255
  const int lane = t & 31;
  const int wv   = t >> 5;           // 0..7

  int chunk = blockIdx.x;
  if (chunk >= kNChunks) return;

  {   // prime buffer 0 (each thread copies 2 edges worth of each array)
    const int e = chunk * kChunk + t * 2;
    async_cp_b64(&s_r[0][t * 2], rows + e);
    async_cp_b64(&s_c[0][t * 2], cols + e);
    async_cp_b64(&s_v[0][t * 2], vals + e);
  }

  int cur = 0;
  for (; chunk < kNChunks; chunk += gridDim.x) {
    wait_async_all();        // my async loads done (LDS data landed)
    __syncthreads();         // everyone's loads visible; prev buffer free

    const int nxt = chunk + gridDim.x;
    if (nxt < kNChunks) {    // stage next chunk into the other buffer
      const int e = nxt * kChunk + t * 2;
      async_cp_b64(&s_r[cur ^ 1][t * 2], rows + e);
      async_cp_b64(&s_c[cur ^ 1][t * 2], cols + e);
      async_cp_b64(&s_v[cur ^ 1][t * 2], vals + e);
    }

    const int eBeg = wv * (kChunk / 8);     // 64 edges per wave
#pragma unroll 8
    for (int k = 0; k < kChunk / 8; ++k) {
      const int   r = s_r[cur][eBeg + k];   // uniform-address LDS broadcast
      const int   c = s_c[cur][eBeg + k];
      const float v = s_v[cur][eBeg + k];
      const float2 xv = *(const float2*)(x + (size_t)c * kD + lane * 2);
      float* yp = y + (size_t)r * kD + lane * 2;
      __hip_atomic_fetch_add(yp + 0, v * xv.x, __ATOMIC_RELAXED,
                             __HIP_MEMORY_SCOPE_AGENT);
      __hip_atomic_fetch_add(yp + 1, v * xv.y, __ATOMIC_RELAXED,
                             __HIP_MEMORY_SCOPE_AGENT);
    }
    __syncthreads();         // done reading buf[cur] before it is re-staged
    cur ^= 1;
  }
}

// ---------------- final mean: acc *= 0.25 via WMMA 16x16 tiles ----------------
typedef __attribute__((ext_vector_type(2))) float v2f;
typedef __attribute__((ext_vector_type(8))) float v8f;

__global__ void __launch_bounds__(256) final_scale_kernel(float* __restrict__ acc) {
  const int gtid = blockIdx.x * blockDim.x + threadIdx.x;
  const int wave = gtid >> 5;
  const int lane = gtid & 31;
  if (wave >= kTiles) return;           // wave-uniform: EXEC stays all-ones

  const int tr = wave >> 2;             // 16-row block (kD/16 == 4 col tiles)
  const int tc = wave & 3;
  float* base = acc + (size_t)tr * 16 * kD + tc * 16;
  const int half = lane >> 4;           // 0: lanes 0-15, 1: lanes 16-31
  const int l    = lane & 15;

#if defined(HAVE_WMMA_F32_4)
  // D = sum_{g=0..3} A_g(16x4) x B_g(4x16) + 0, with B = 0.25*I16 sliced in K.
  // A layout (32-bit A 16x4): lanes 0-15 hold K={4g,4g+1}, lanes 16-31 K={4g+2,4g+3}.
  v8f dacc = {};
  const float* arow = base + (size_t)l * kD;
#pragma unroll
  for (int g = 0; g < 4; ++g) {
    const int k0 = 4 * g + half * 2;
    const float2 a2 = *(const float2*)(arow + k0);
    v2f A; A.x = a2.x; A.y = a2.y;
    v2f B;                                // B_g[k][n] = 0.25 * (n == 4g+k)
    B.x = (l == k0)     ? 0.25f : 0.0f;
    B.y = (l == k0 + 1) ? 0.25f : 0.0f;
    dacc = __builtin_amdgcn_wmma_f32_16x16x4_f32(
        /*neg_a=*/false, A, /*neg_b=*/false, B,
        /*c_mod=*/(short)0, dacc, /*reuse_a=*/false, /*reuse_b=*/false);
  }
  // C/D layout: VGPR j -> M = j + 8*half, N = l.
#pragma unroll
  for (int j = 0; j < 8; ++j)
    base[(size_t)(half * 8 + j) * kD + l] = dacc[j];
#else
#pragma unroll
  for (int j = 0; j < 8; ++j) {
    const size_t idx = (size_t)(half * 8 + j) * kD + l;
    base[idx] *= 0.25f;
  }
#endif
}

// ---------------- host side ----------------
extern "C" void kernel_launch(void* const* d_in, const int* in_sizes, int n_in,
                              void* d_out, int out_size, void* d_ws, size_t ws_size,
                              hipStream_t stream) {
  const float* ue   = (const float*)d_in[0];  // user_emb  [100000,64]
  const float* ie   = (const float*)d_in[1];  // item_emb  [50000,64]
  const float* vals = (const float*)d_in[2];  // adj_vals  [4.8M]
  const int*   rows = (const int*)  d_in[3];  // adj_rows  [4.8M]
  const int*   cols = (const int*)  d_in[4];  // adj_cols  [4.8M]

  float* acc = (float*)d_out;                 // running sum, N*D floats
  float* x0  = (float*)d_ws;                  // ping
  float* x1  = x0 + (size_t)kElems;           // pong  (needs 76.8 MB of ws)

  const int ewBlocks = 2048;                  // grid-stride elementwise grids

  init_kernel<<<ewBlocks, 256, 0, stream>>>((const float4*)ue, (const float4*)ie,
                                            (float4*)x0, (float4*)acc);

  float* xc = x0;
  float* xn = x1;
  for (int layer = 0; layer < kLayers; ++layer) {
    zero_kernel<<<ewBlocks, 256, 0, stream>>>((float4*)xn);
    spmm_kernel<<<1875, 256, 0, stream>>>(rows, cols, vals, xc, xn);
    acc_add_kernel<<<ewBlocks, 256, 0, stream>>>((float4*)acc, (const float4*)xn);
    float* tmp = xc; xc = xn; xn = tmp;
  }

  final_scale_kernel<<<(kTiles * 32 + 255) / 256, 256, 0, stream>>>(acc);
}